// stencilCNN_4887672783587
// MI455X (gfx1250) — compile-verified
//
#include <hip/hip_runtime.h>

// Fused WENO5 + RK4, periodic last axis. One HBM pass (64 MB traffic, ~2.8us
// floor at 23.3 TB/s) but compute-bound: ~46 VALU + 4 v_rcp_f32 per interface
// flux, ~4 fluxes per output point. Hot loops process 2 interfaces/lane via
// 2-wide f32 vectors -> CDNA5 V_PK_{FMA,ADD,MUL}_F32 packed VALU.
// CDNA5 async path: global_load_async_to_lds_b128 + s_wait_asynccnt staging.

#define TILE      4096
#define HALO      16                    // need 12 left / 8 right
#define NTHREADS  256
#define PTS       (TILE / NTHREADS)     // 16 outputs per thread
#define LDSN      (TILE + 2 * HALO)     // 4128 floats per LDS array
#define EPS_W     1e-6f
#define INV_DX    100.0f                // 1/0.01

typedef float v2f __attribute__((ext_vector_type(2)));

__device__ __forceinline__ v2f v2rcp(v2f x) {
  v2f r;
  r.x = __builtin_amdgcn_rcpf(x.x);
  r.y = __builtin_amdgcn_rcpf(x.y);
  return r;
}

// Two adjacent interface fluxes kk[j], kk[j+1] from 6 consecutive inputs
// f[-3..+2] (f0 == src[j]).
__device__ __forceinline__ v2f weno_flux2(
    float fm3, float fm2, float fm1, float f0, float fp1, float fp2,
    float cA0, float cA1, float cA2,
    float cB1, float cB2, float cB3,
    float cC2, float cC3, float cC4) {
  v2f a, b, c, d, e;
  a.x = fm3; a.y = fm2;
  b.x = fm2; b.y = fm1;
  c.x = fm1; c.y = f0;
  d.x = f0;  d.y = fp1;
  e.x = fp1; e.y = fp2;
  v2f p0 = cA0 * a + cA1 * b + cA2 * c;
  v2f p1 = cB1 * b + cB2 * c + cB3 * d;
  v2f p2 = cC2 * c + cC3 * d + cC4 * e;
  v2f t0 = a - 2.0f * b + c;
  v2f t1 = a - 4.0f * b + 3.0f * c;
  v2f s0 = (13.0f / 12.0f) * t0 * t0 + 0.25f * t1 * t1;
  v2f t2 = b - 2.0f * c + d;
  v2f t3 = b - d;
  v2f s1 = (13.0f / 12.0f) * t2 * t2 + 0.25f * t3 * t3;
  v2f t4 = c - 2.0f * d + e;
  v2f t5 = 3.0f * c - 4.0f * d + e;
  v2f s2 = (13.0f / 12.0f) * t4 * t4 + 0.25f * t5 * t5;
  v2f r0 = v2rcp(EPS_W + s0);
  v2f r1 = v2rcp(EPS_W + s1);
  v2f r2 = v2rcp(EPS_W + s2);
  v2f a0 = 0.1f * r0 * r0;
  v2f a1 = 0.6f * r1 * r1;
  v2f a2 = 0.3f * r2 * r2;
  return (a0 * p0 + a1 * p1 + a2 * p2) * v2rcp(a0 + a1 + a2);
}

__global__ __launch_bounds__(NTHREADS) void weno_rk4_fused_4887672783587(
    const float* __restrict__ u, const float* __restrict__ stencil,
    const float* __restrict__ k_coef, const float* __restrict__ fc_w,
    float* __restrict__ out, int N) {
  __shared__ __align__(16) float sU[LDSN];   // original u tile (+halo)
  __shared__ __align__(16) float sV[LDSN];   // stage input u + c*k_prev
  __shared__ __align__(16) float sKK[LDSN];  // interface fluxes kk[j]

  const int tid = threadIdx.x;
  const long long rowBase = (long long)blockIdx.y * (long long)N;
  const int tileStart = blockIdx.x * TILE;

  // uniform coefficients (scalar loads)
  const float st0 = stencil[0], st1 = stencil[1], st2 = stencil[2];
  const float cA0 = st0, cA1 = -0.5f - 2.0f * st0, cA2 = 1.5f + st0;
  const float cB1 = st1, cB2 = 0.5f - 2.0f * st1, cB3 = 0.5f + st1;
  const float cC2 = st2, cC3 = 1.5f - 2.0f * st2, cC4 = -0.5f + st2;
  const float cS[3] = {k_coef[0], k_coef[1], k_coef[2]};
  const float wS[4] = {fc_w[0], fc_w[1], fc_w[2], fc_w[3]};

  // ---- async tile staging: global -> LDS (ASYNCcnt path) ----
  // Region [tileStart-16, tileStart+TILE+16). Chunk starts and N are
  // multiples of 16 floats, so no 64B chunk straddles the wrap boundary and
  // all transfers stay 16B aligned. offset: applies to both LDS and global.
  {
    const unsigned ldsBase = (unsigned)(unsigned long long)(void*)&sU[0];
    const float* rowPtr = u + rowBase;
    {
      int f = tileStart - HALO + 16 * tid;          // 64B per thread (bulk)
      if (f < 0) f += N; else if (f >= N) f -= N;
      unsigned long long ga = (unsigned long long)(const void*)(rowPtr + f);
      unsigned la = ldsBase + 64u * (unsigned)tid;
      asm volatile("global_load_async_to_lds_b128 %0, %1, off"           :: "v"(la), "v"(ga) : "memory");
      asm volatile("global_load_async_to_lds_b128 %0, %1, off offset:16" :: "v"(la), "v"(ga) : "memory");
      asm volatile("global_load_async_to_lds_b128 %0, %1, off offset:32" :: "v"(la), "v"(ga) : "memory");
      asm volatile("global_load_async_to_lds_b128 %0, %1, off offset:48" :: "v"(la), "v"(ga) : "memory");
    }
    if (tid < 8) {                                   // last 32 floats of region
      int f = tileStart - HALO + 4096 + 4 * tid;
      if (f < 0) f += N; else if (f >= N) f -= N;
      unsigned long long ga = (unsigned long long)(const void*)(rowPtr + f);
      unsigned la = ldsBase + 4u * (4096u + 4u * (unsigned)tid);
      asm volatile("global_load_async_to_lds_b128 %0, %1, off" :: "v"(la), "v"(ga) : "memory");
    }
    asm volatile("s_wait_asynccnt 0x0" ::: "memory");
  }
  __syncthreads();

  v2f du2[PTS / 2];
#pragma unroll
  for (int r = 0; r < PTS / 2; ++r) { du2[r].x = 0.0f; du2[r].y = 0.0f; }

  // Needed interface ranges (inclusive, tile-relative):
  //   s0: [-9,4102]  s1: [-6,4100]  s2: [-3,4098]  s3: [0,4096]
  // Pair bases rounded to even; the few extra interfaces computed for
  // alignment read in-bounds, possibly stale or unwritten LDS and are never consumed.
  const int kkPairLo[4]  = {-10, -6, -4, 0};
  const int kkPairCnt[4] = {2057, 2054, 2052, 2049};
  // Stage-input update ranges: s0: [-9,4101]  s1: [-6,4099]  s2: [-3,4097]
  const int vPairLo[3]   = {-10, -6, -4};
  const int vPairCnt[3]  = {2056, 2053, 2051};

  float* sp = sU;  // stage-1 input is u itself
  for (int s = 0; s < 4; ++s) {
    // 1) two interface fluxes per lane-iteration into sKK (even base -> b64)
    const int lo = kkPairLo[s], n = kkPairCnt[s];
    for (int idx = tid; idx < n; idx += NTHREADS) {
      const int j = HALO + lo + 2 * idx;
      v2f kk = weno_flux2(sp[j - 3], sp[j - 2], sp[j - 1],
                          sp[j], sp[j + 1], sp[j + 2],
                          cA0, cA1, cA2, cB1, cB2, cB3, cC2, cC3, cC4);
      *(v2f*)&sKK[j] = kk;
    }
    __syncthreads();

    // 2) accumulate du for this thread's owned output pairs
    const float ws = wS[s];
#pragma unroll
    for (int r = 0; r < PTS / 2; ++r) {
      const int i = HALO + 2 * tid + r * (2 * NTHREADS);
      const float k0 = sKK[i], k1 = sKK[i + 1], k2 = sKK[i + 2];
      v2f klo, khi;
      klo.x = k0; klo.y = k1;
      khi.x = k1; khi.y = k2;
      v2f k = (khi - klo) * INV_DX;
      du2[r] += ws * k;
    }

    // 3) next stage input V = U + c*k over its halo range (pairwise)
    if (s < 3) {
      const float cc = cS[s];
      const int l2 = vPairLo[s], n2 = vPairCnt[s];
      for (int idx = tid; idx < n2; idx += NTHREADS) {
        const int i = HALO + l2 + 2 * idx;
        const float k0 = sKK[i], k1 = sKK[i + 1], k2 = sKK[i + 2];
        v2f klo, khi;
        klo.x = k0; klo.y = k1;
        khi.x = k1; khi.y = k2;
        v2f k = (khi - klo) * INV_DX;
        v2f uu = *(const v2f*)&sU[i];
        *(v2f*)&sV[i] = uu + cc * k;
      }
      sp = sV;
    }
    __syncthreads();  // protect sKK/sV before next stage rewrites them
  }

  float* o = out + rowBase + tileStart;
#pragma unroll
  for (int r = 0; r < PTS / 2; ++r) {
    *(v2f*)&o[2 * tid + r * (2 * NTHREADS)] = du2[r];
  }
}

extern "C" void kernel_launch(void* const* d_in, const int* in_sizes, int n_in,
                              void* d_out, int out_size, void* d_ws, size_t ws_size,
                              hipStream_t stream) {
  const float* u       = (const float*)d_in[0];
  const float* stencil = (const float*)d_in[1];
  const float* k_coef  = (const float*)d_in[2];
  const float* fc_w    = (const float*)d_in[3];
  float* out = (float*)d_out;

  const int N = 524288;                 // last-axis length (reference)
  const int B = in_sizes[0] / N;        // 16 rows (B*C)
  dim3 grid(N / TILE, B);               // 128 x 16 = 2048 blocks
  weno_rk4_fused_4887672783587<<<grid, NTHREADS, 0, stream>>>(
      u, stencil, k_coef, fc_w, out, N);
}